// MultiHeadSelfAttention_44719199486679
// MI455X (gfx1250) — compile-verified
//
#include <hip/hip_runtime.h>

// ---------------------------------------------------------------------------
// MultiHeadSelfAttention on gfx1250 (MI455X), fp32 WMMA path.
// B=8, C=256, H=W=32 (N=1024), heads=8, head_dim=32.
// ---------------------------------------------------------------------------

typedef float v2f __attribute__((ext_vector_type(2)));
typedef float v4f __attribute__((ext_vector_type(4)));
typedef float v8f __attribute__((ext_vector_type(8)));

#define BATCH 8
#define CH    256
#define NPIX  1024
#define NH    8
#define HD    32

static __device__ __forceinline__ v8f wmma_f32(v2f a, v2f b, v8f c) {
    // V_WMMA_F32_16X16X4_F32 : D(16x16 f32) = A(16x4 f32) x B(4x16 f32) + C
    return __builtin_amdgcn_wmma_f32_16x16x4_f32(false, a, false, b,
                                                 (short)0, c, false, false);
}

// ---------------------------------------------------------------------------
// Kernel 1: fused Q/K/V projection.
//   y[o,n] = sum_c W[o,c] * x[b,c,n] + bias[o]
// grid = B * 3proj * 16 Mtiles * 8 Ngroups ; block = 256 (8 waves)
// Each wave computes one 16x16 output tile with 64 fp32 WMMAs (K=256),
// split across two accumulators to break the D->C dependency chain.
// Q,K stored [b][c][n]; V stored transposed per head: Vt[b][h][n][d].
// ---------------------------------------------------------------------------
__global__ void mhsa_qkv_proj(const float* __restrict__ x,
                              const float* __restrict__ Wq, const float* __restrict__ bq,
                              const float* __restrict__ Wk, const float* __restrict__ bk,
                              const float* __restrict__ Wv, const float* __restrict__ bv,
                              float* __restrict__ Qb, float* __restrict__ Kb,
                              float* __restrict__ Vt)
{
    const int lane = threadIdx.x & 31;
    const int wave = threadIdx.x >> 5;
    const int half = lane >> 4;       // 0: lanes 0-15, 1: lanes 16-31
    const int l15  = lane & 15;
    const int kb   = half * 2;        // A/B K sub-offset per half-wave

    int id = blockIdx.x;
    const int ng = id & 7;  id >>= 3;
    const int mt = id & 15; id >>= 4;
    const int p  = id % 3;            // 0=Q 1=K 2=V
    const int b  = id / 3;

    const int m0 = mt * 16;
    const int n0 = (ng * 8 + wave) * 16;

    const float* W;
    const float* bias;
    if      (p == 0) { W = Wq; bias = bq; }
    else if (p == 1) { W = Wk; bias = bk; }
    else             { W = Wv; bias = bv; }

    // acc0 preloaded with bias (D layout: VGPR j -> row m0+j (+8 upper half))
    v8f acc0, acc1 = {0.f,0.f,0.f,0.f,0.f,0.f,0.f,0.f};
#pragma unroll
    for (int j = 0; j < 8; ++j) acc0[j] = bias[m0 + j + half * 8];

    const float* xb = x + (size_t)b * CH * NPIX;

    for (int k0 = 0; k0 < CH; k0 += 8) {
        // A: W tile 16x4, per-lane contiguous float2 (K+0,K+1 | K+2,K+3)
        v2f a0 = *(const v2f*)(W + (m0 + l15) * CH + k0 + kb);
        v2f a1 = *(const v2f*)(W + (m0 + l15) * CH + k0 + 4 + kb);
        // B: x tiles 4x16, lane-contiguous along n
        v2f b0, b1;
        b0.x = xb[(size_t)(k0 + kb        ) * NPIX + n0 + l15];
        b0.y = xb[(size_t)(k0 + kb + 1    ) * NPIX + n0 + l15];
        b1.x = xb[(size_t)(k0 + 4 + kb    ) * NPIX + n0 + l15];
        b1.y = xb[(size_t)(k0 + 4 + kb + 1) * NPIX + n0 + l15];
        acc0 = wmma_f32(a0, b0, acc0);
        acc1 = wmma_f32(a1, b1, acc1);
    }
#pragma unroll
    for (int j = 0; j < 8; ++j) acc0[j] += acc1[j];

    if (p < 2) {
        float* dst = (p == 0 ? Qb : Kb) + (size_t)b * CH * NPIX;
#pragma unroll
        for (int j = 0; j < 8; ++j)
            dst[(size_t)(m0 + j + half * 8) * NPIX + n0 + l15] = acc0[j];
    } else {
        // transpose-store V into Vt[b][h][n][d]
#pragma unroll
        for (int j = 0; j < 8; ++j) {
            int o  = m0 + j + half * 8;
            int hh = o >> 5, d = o & 31;
            Vt[((size_t)(b * NH + hh) * NPIX + n0 + l15) * HD + d] = acc0[j];
        }
    }
}

// ---------------------------------------------------------------------------
// Kernel 2: attention for one (b, h, 64-query block).
// grid = B*NH*(NPIX/64) = 1024 ; block = 256 (8 waves)
// LDS: 64 x 1028 padded fp32 score rows (stride 1028 == 4 mod 64 banks
//      -> conflict-free ds_load_b64 of the A operand in phase C) + reductions.
// ---------------------------------------------------------------------------
#define QBLK  64
#define RSTR  1028   // padded LDS row stride in floats

__global__ void mhsa_attn(const float* __restrict__ Qb,
                          const float* __restrict__ Kb,
                          const float* __restrict__ Vt,
                          const float* __restrict__ x,
                          const float* __restrict__ gamma,
                          float* __restrict__ out)
{
    extern __shared__ float smem[];
    float* sc   = smem;                 // QBLK * RSTR score/prob rows
    float* redA = smem + QBLK * RSTR;   // 256 partial maxima
    float* redB = redA + 256;           // 256 partial sums

    const int lane = threadIdx.x & 31;
    const int wave = threadIdx.x >> 5;
    const int half = lane >> 4;
    const int l15  = lane & 15;
    const int kb   = half * 2;

    int id = blockIdx.x;
    const int qb = id & 15;
    const int h  = (id >> 4) & 7;
    const int b  = id >> 7;
    const int q0 = qb * QBLK;

    const float* Qh = Qb + (size_t)(b * CH + h * HD) * NPIX;
    const float* Kh = Kb + (size_t)(b * CH + h * HD) * NPIX;
    const float* Vh = Vt + (size_t)(b * NH + h) * NPIX * HD;

    // ---------------- Phase A: S = q^T k / sqrt(hd) -> LDS -----------------
    // Wave owns a fixed q-subtile (qs = wave&3) and half the kt range:
    // hoist the 16x32 Q tile (8 K-chunks) into registers once.
    {
        const int qs  = wave & 3;                  // 0..3
        const int kt0 = (wave >> 2) * 32;          // 0 or 32
        v2f aq[8];
#pragma unroll
        for (int dc = 0; dc < 8; ++dc) {
            aq[dc].x = Qh[(size_t)(dc * 4 + kb    ) * NPIX + q0 + qs * 16 + l15];
            aq[dc].y = Qh[(size_t)(dc * 4 + kb + 1) * NPIX + q0 + qs * 16 + l15];
        }
        const float s = 0.17677669529663687f;      // 1/sqrt(32)
        for (int kt = kt0; kt < kt0 + 32; ++kt) {
            // prefetch next K tile's first rows
            __builtin_prefetch(Kh + (size_t)kb * NPIX + (kt + 1) * 16 + l15, 0, 1);
            v8f acc0 = {0.f,0.f,0.f,0.f,0.f,0.f,0.f,0.f};
            v8f acc1 = {0.f,0.f,0.f,0.f,0.f,0.f,0.f,0.f};
#pragma unroll
            for (int dc = 0; dc < 8; dc += 2) {
                v2f b0, b1;
                b0.x = Kh[(size_t)(dc * 4 + kb        ) * NPIX + kt * 16 + l15];
                b0.y = Kh[(size_t)(dc * 4 + kb + 1    ) * NPIX + kt * 16 + l15];
                b1.x = Kh[(size_t)(dc * 4 + 4 + kb    ) * NPIX + kt * 16 + l15];
                b1.y = Kh[(size_t)(dc * 4 + 4 + kb + 1) * NPIX + kt * 16 + l15];
                acc0 = wmma_f32(aq[dc],     b0, acc0);
                acc1 = wmma_f32(aq[dc + 1], b1, acc1);
            }
#pragma unroll
            for (int j = 0; j < 8; ++j)
                sc[(qs * 16 + j + half * 8) * RSTR + kt * 16 + l15]
                    = (acc0[j] + acc1[j]) * s;
        }
    }
    __syncthreads();

    // ---------------- Phase B: row softmax (64 rows x 4 threads) -----------
    // Each thread owns a contiguous, 16B-aligned 256-float segment -> float4.
    {
        const int row = threadIdx.x >> 2;
        const int seg = threadIdx.x & 3;
        v4f* r4 = (v4f*)(sc + row * RSTR + seg * 256);

        float m = -3.402823466e38f;
        for (int i = 0; i < 64; ++i) {
            v4f v = r4[i];
            m = fmaxf(m, fmaxf(fmaxf(v[0], v[1]), fmaxf(v[2], v[3])));
        }
        redA[threadIdx.x] = m;                     // tid == row*4+seg
        __syncthreads();

        const float mr = fmaxf(fmaxf(redA[row * 4 + 0], redA[row * 4 + 1]),
                               fmaxf(redA[row * 4 + 2], redA[row * 4 + 3]));
        float ssum = 0.f;
        for (int i = 0; i < 64; ++i) {
            v4f v = r4[i];
            v4f e;
#pragma unroll
            for (int c = 0; c < 4; ++c) e[c] = __expf(v[c] - mr);
            r4[i] = e;
            ssum += e[0] + e[1] + e[2] + e[3];
        }
        redB[threadIdx.x] = ssum;
        __syncthreads();

        const float tot = redB[row * 4 + 0] + redB[row * 4 + 1] +
                          redB[row * 4 + 2] + redB[row * 4 + 3];
        const float inv = 1.f / tot;
        for (int i = 0; i < 64; ++i) {
            v4f v = r4[i];
#pragma unroll
            for (int c = 0; c < 4; ++c) v[c] *= inv;
            r4[i] = v;
        }
    }
    __syncthreads();

    // ---------------- Phase C: O = P @ V^T, fused gamma*O + x --------------
    {
        const int qs = wave >> 1;                  // 0..3
        const int dt = wave & 1;                   // 0..1
        v8f acc0 = {0.f,0.f,0.f,0.f,0.f,0.f,0.f,0.f};
        v8f acc1 = {0.f,0.f,0.f,0.f,0.f,0.f,0.f,0.f};
        for (int m0 = 0; m0 < NPIX; m0 += 8) {
            __builtin_prefetch(Vh + (size_t)(m0 + 16) * HD + dt * 16 + l15, 0, 1);
            // A: probs from LDS, contiguous float2 per lane (conflict-free)
            v2f a0 = *(const v2f*)(sc + (qs * 16 + l15) * RSTR + m0 + kb);
            v2f a1 = *(const v2f*)(sc + (qs * 16 + l15) * RSTR + m0 + 4 + kb);
            // B: Vt rows, lane-contiguous along d
            v2f b0, b1;
            b0.x = Vh[(size_t)(m0 + kb        ) * HD + dt * 16 + l15];
            b0.y = Vh[(size_t)(m0 + kb + 1    ) * HD + dt * 16 + l15];
            b1.x = Vh[(size_t)(m0 + 4 + kb    ) * HD + dt * 16 + l15];
            b1.y = Vh[(size_t)(m0 + 4 + kb + 1) * HD + dt * 16 + l15];
            acc0 = wmma_f32(a0, b0, acc0);
            acc1 = wmma_f32(a1, b1, acc1);
        }
        const float g = gamma[0];
#pragma unroll
        for (int j = 0; j < 8; ++j) {
            const int n = q0 + qs * 16 + j + half * 8;
            const int c = h * HD + dt * 16 + l15;
            const size_t idx = (size_t)(b * CH + c) * NPIX + n;
            out[idx] = g * (acc0[j] + acc1[j]) + x[idx];
        }
    }
}

// ---------------------------------------------------------------------------
extern "C" void kernel_launch(void* const* d_in, const int* in_sizes, int n_in,
                              void* d_out, int out_size, void* d_ws, size_t ws_size,
                              hipStream_t stream)
{
    const float* x     = (const float*)d_in[0];
    const float* Wq    = (const float*)d_in[1];
    const float* bq    = (const float*)d_in[2];
    const float* Wk    = (const float*)d_in[3];
    const float* bk    = (const float*)d_in[4];
    const float* Wv    = (const float*)d_in[5];
    const float* bv    = (const float*)d_in[6];
    const float* gamma = (const float*)d_in[7];
    float* out = (float*)d_out;

    float* ws = (float*)d_ws;
    const size_t plane = (size_t)BATCH * CH * NPIX;   // 2M floats = 8 MB
    float* Q  = ws;
    float* K  = ws + plane;
    float* Vt = ws + 2 * plane;                       // 24 MB total scratch

    // Projection: 8 batches * 3 proj * 16 Mtiles * 8 Ngroups blocks
    mhsa_qkv_proj<<<BATCH * 3 * 16 * 8, 256, 0, stream>>>(
        x, Wq, bq, Wk, bk, Wv, bv, Q, K, Vt);

    // Attention: 1024 blocks, 256 threads, ~259 KB dynamic LDS per workgroup
    const size_t lds_bytes = (size_t)(QBLK * RSTR + 512) * sizeof(float);
    mhsa_attn<<<BATCH * NH * (NPIX / QBLK), 256, lds_bytes, stream>>>(
        Q, K, Vt, x, gamma, out);
}